// RelativeDepthLoss_87359634801366
// MI455X (gfx1250) — compile-verified
//
#include <hip/hip_runtime.h>
#include <hip/hip_bf16.h>
#include <stdint.h>

typedef __attribute__((ext_vector_type(2))) float v2f;
typedef __attribute__((ext_vector_type(8))) float v8f;

#define HH 64
#define WW 64
#define NN 4096           // H*W
#define NPAIR (4096*4096) // 16,777,216 pairs

// ---------------- Kernel A: Sobel gradient loss numerator + mask count -------
__global__ void rd_grad_kernel(const float* __restrict__ pred,
                               const float* __restrict__ gt,
                               const unsigned char* __restrict__ mask,
                               float* __restrict__ ws) {
    __shared__ float e[NN];
    __shared__ float redn[256];
    __shared__ float redc[256];
    const int tid = threadIdx.x;
    for (int k = tid; k < NN; k += 256) e[k] = pred[k] - gt[k];
    __syncthreads();

    float num = 0.f, cnt = 0.f;
    for (int k = tid; k < NN; k += 256) {
        const int y = k >> 6, x = k & 63;
        float n[3][3];
#pragma unroll
        for (int dy = 0; dy < 3; ++dy)
#pragma unroll
            for (int dx = 0; dx < 3; ++dx) {
                const int yy = y + dy - 1, xx = x + dx - 1;
                n[dy][dx] = (yy >= 0 && yy < HH && xx >= 0 && xx < WW)
                                ? e[(yy << 6) + xx] : 0.f;
            }
        // cross-correlation (XLA/torch style), zero padding; sobel is linear so
        // gx_p-gx_g = sobel_x(pred-gt)
        const float gx = (n[0][2] - n[0][0]) + 2.f * (n[1][2] - n[1][0]) + (n[2][2] - n[2][0]);
        const float gy = (n[2][0] + 2.f * n[2][1] + n[2][2]) - (n[0][0] + 2.f * n[0][1] + n[0][2]);
        const float m = mask[k] ? 1.f : 0.f;
        num += (fabsf(gx) + fabsf(gy)) * m;
        cnt += m;
    }
    redn[tid] = num; redc[tid] = cnt;
    __syncthreads();
    for (int s = 128; s > 0; s >>= 1) {
        if (tid < s) { redn[tid] += redn[tid + s]; redc[tid] += redc[tid + s]; }
        __syncthreads();
    }
    if (tid == 0) { ws[256] = redn[0]; ws[257] = redc[0]; }
}

// ---------------- Kernel B: O(N^2) pairwise loss, WMMA-accumulated -----------
// 256 blocks x 256 threads (8 wave32/block) -> 2048 waves; each wave owns two
// full rows i (all 4096 j's each). Row-invariants (a_i, row/col of i, dx^2 per
// lane) are hoisted; (a_j, m_j) pairs come from LDS as one 16B load per 2 pairs.
// Rows with m_i == 0 are skipped with a wave-uniform branch (EXEC stays all-1s,
// so WMMA remains legal). 64 pair-values per V_WMMA_F32_16X16X4_F32 are reduced
// through an all-ones B matrix (row sums replicated x16 across columns).
__global__ void rd_pair_kernel(const float* __restrict__ pred,
                               const float* __restrict__ gt,
                               const unsigned char* __restrict__ mask,
                               float* __restrict__ ws) {
    __shared__ float2 am_sh[NN];   // interleaved (a_k, m_k)
    __shared__ float red[256];
    const int tid = threadIdx.x;
    for (int k = tid; k < NN; k += 256) {
        am_sh[k] = make_float2(pred[k] - gt[k], mask[k] ? 1.f : 0.f);
    }
    __syncthreads();

    const int lane   = tid & 31;
    const int waveId = (int)blockIdx.x * 8 + (tid >> 5);   // 0..2047

    v8f acc = {};
    const v2f bones = {1.f, 1.f};  // all-ones B (4x16): D[m][n] = rowsum(A) + C
    const int jcol0 = lane * 2;    // j & 63 for slot 0 -- per-lane constant

#pragma unroll
    for (int ii = 0; ii < 2; ++ii) {
        const int i = waveId * 2 + ii;                     // row index 0..4095
        const float2 ami = am_sh[i];                       // wave-uniform load
        if (ami.y != 0.f) {                                // m_i==1 (skip 0-rows)
            const float ai   = ami.x;
            const int   rowi = i >> 6;
            const int   coli = i & 63;
            const float dxl0 = (float)(coli - jcol0);
            const float dxl1 = (float)(coli - (jcol0 + 1));
            const float dx2_0 = dxl0 * dxl0;               // hoisted for all 64 iters
            const float dx2_1 = dxl1 * dxl1;
            for (int rowj = 0; rowj < 64; ++rowj) {
                const float sdy  = (float)(rowi - rowj);
                const float sdy2 = sdy * sdy;
                const float4 am = *reinterpret_cast<const float4*>(
                    am_sh + ((rowj << 6) + jcol0));        // (a_j0,m_j0,a_j1,m_j1), 16B aligned
                const float inv0 =
                    __builtin_amdgcn_rcpf(__builtin_amdgcn_sqrtf(sdy2 + dx2_0) + 0.001f);
                const float inv1 =
                    __builtin_amdgcn_rcpf(__builtin_amdgcn_sqrtf(sdy2 + dx2_1) + 0.001f);
                const float v0 = fabsf(ai - am.x) * am.y * inv0;
                const float v1 = fabsf(ai - am.z) * am.w * inv1;
                v2f av = {v0, v1};
                // D(16x16,f32) = A(16x4,f32) x ones(4x16) + C
                acc = __builtin_amdgcn_wmma_f32_16x16x4_f32(
                    /*neg_a=*/false, av, /*neg_b=*/false, bones,
                    /*c_mod=*/(short)0, acc, /*reuse_a=*/false, /*reuse_b=*/false);
            }
        }
    }

    float s = acc[0] + acc[1] + acc[2] + acc[3] + acc[4] + acc[5] + acc[6] + acc[7];
    red[tid] = s;
    __syncthreads();
    for (int st = 128; st > 0; st >>= 1) {
        if (tid < st) red[tid] += red[tid + st];
        __syncthreads();
    }
    // each pair value was counted 16x (replication across the 16 columns)
    if (tid == 0) ws[blockIdx.x] = red[0] * (1.0f / 16.0f);
}

// ---------------- Kernel C: deterministic final combine ----------------------
__global__ void rd_final_kernel(const float* __restrict__ ws, float* __restrict__ out) {
    __shared__ float red[256];
    const int tid = threadIdx.x;
    red[tid] = ws[tid];               // 256 per-block pairwise partials
    __syncthreads();
    for (int s = 128; s > 0; s >>= 1) {
        if (tid < s) red[tid] += red[tid + s];
        __syncthreads();
    }
    if (tid == 0) {
        const float pair_sum  = red[0];
        const float grad_num  = ws[256];
        const float cnt       = ws[257];
        const float grad_loss = grad_num / cnt;
        // pair_mask.sum() == cnt^2 exactly
        const float denom     = __builtin_amdgcn_sqrtf(cnt * cnt + 1e-5f);
        const float pair_loss = pair_sum / denom;
        const float total     = 0.5f * grad_loss + 0.5f * pair_loss; // WEIGHT = 0.5
        out[0] = logf(total + 1.0f);
    }
}

extern "C" void kernel_launch(void* const* d_in, const int* in_sizes, int n_in,
                              void* d_out, int out_size, void* d_ws, size_t ws_size,
                              hipStream_t stream) {
    (void)in_sizes; (void)n_in; (void)out_size; (void)ws_size;
    const float*         pred = (const float*)d_in[0];
    const float*         gt   = (const float*)d_in[1];
    const unsigned char* mask = (const unsigned char*)d_in[2]; // jax bool = 1 byte
    float* out = (float*)d_out;
    float* ws  = (float*)d_ws;   // [0..255] pair partials, [256] grad_num, [257] cnt

    rd_grad_kernel<<<1,   256, 0, stream>>>(pred, gt, mask, ws);
    rd_pair_kernel<<<256, 256, 0, stream>>>(pred, gt, mask, ws);
    rd_final_kernel<<<1,  256, 0, stream>>>(ws, out);
}